// Quantize_22162031247919
// MI455X (gfx1250) — compile-verified
//
#include <hip/hip_runtime.h>

// ---------------------------------------------------------------------------
// VQ-VAE vector quantization for MI455X (gfx1250, wave32).
//   samples      : [16384, 128]  f32   (d_in[0])
//   cluster_mean : [128, 10000]  f32   (d_in[1])
//   d_out layout : quantize[16384*128] | cluster_index[16384] (as f32) | loss[1]
//
// Distance GEMM on V_WMMA_F32_16X16X4_F32 (full fp32 matrix path -> exact
// argmin fidelity). Code dimension split into 5 chunks (625 tiles = 5 x 125)
// for WGP fill (1280 workgroups / 5120 waves). B fragments are preloaded into
// a register array each tile so the DS->WMMA pipeline is deep by construction.
// ---------------------------------------------------------------------------

typedef __attribute__((ext_vector_type(2))) float v2f;
typedef __attribute__((ext_vector_type(4))) float v4f;
typedef __attribute__((ext_vector_type(8))) float v8f;

#define DIMS     128
#define KCODES   10000
#define NSAMP    16384
#define CHUNKS   5
#define CTILES   125                  // code tiles of 16 per chunk
#define CODES_PER_CHUNK (CTILES * 16) // 2000
#define QSIZE    (NSAMP * DIMS)       // 2097152
#define IDX_OFF  QSIZE
#define LOSS_OFF (QSIZE + NSAMP)
#define BROW     20                   // padded LDS row stride (floats), 16B aligned

// ---------------- codebook norms: ||c_k||^2 ----------------
__global__ __launch_bounds__(256) void cnorm_kernel(const float* __restrict__ CM,
                                                    float* __restrict__ cn) {
    int k = blockIdx.x * 256 + threadIdx.x;
    if (k < KCODES) {
        float s = 0.f;
        #pragma unroll 8
        for (int d = 0; d < DIMS; ++d) {
            float v = CM[(size_t)d * KCODES + k];   // coalesced across threads
            s += v * v;
        }
        cn[k] = s;
    }
}

// ---------------- distance + per-chunk argmin ----------------
// grid (256, 5) x block 128 (4 waves). Wave w owns samples m0+16w .. +15;
// block scans chunk blockIdx.y (125 tiles of 16 codes).
__global__ __launch_bounds__(128) void vq_dist_kernel(const float* __restrict__ S,
                                                      const float* __restrict__ CM,
                                                      const float* __restrict__ cnorm,
                                                      float* __restrict__ minV,
                                                      int* __restrict__ minI) {
    __shared__ float Bs[2][DIMS][BROW];   // double-buffered 128x16 code tile
    __shared__ float redV[4][8][32];
    __shared__ int   redI[4][8][32];

    const int tid   = threadIdx.x;
    const int w     = tid >> 5;      // wave 0..3
    const int lane  = tid & 31;
    const int half  = lane >> 4;     // 0: K slots {0,1}, 1: K slots {2,3}
    const int lc    = lane & 15;     // row-of-A / col-of-B id
    const int m0    = blockIdx.x * 64;
    const int chunk = blockIdx.y;
    const int nbase = chunk * CODES_PER_CHUNK;
    const int mrow  = m0 + w * 16 + lc;

    // ---- preload this lane's A fragments for all of D (once) ----
    v2f a[32];
    const float* srow = S + (size_t)mrow * DIMS + half * 2;
    #pragma unroll
    for (int kk = 0; kk < 32; ++kk)
        a[kk] = *(const v2f*)(srow + kk * 4);

    // ---- cooperative B tile staging (thread t -> codebook row t) ----
    v4f regs[4];
    const float* browp = CM + (size_t)tid * KCODES + nbase;
    #define LOAD_B(n0)                                                 \
        { const v4f* p = (const v4f*)(browp + (n0));                   \
          regs[0] = p[0]; regs[1] = p[1]; regs[2] = p[2]; regs[3] = p[3]; }
    #define STORE_B(buf)                                               \
        { *(v4f*)&Bs[buf][tid][0]  = regs[0];                          \
          *(v4f*)&Bs[buf][tid][4]  = regs[1];                          \
          *(v4f*)&Bs[buf][tid][8]  = regs[2];                          \
          *(v4f*)&Bs[buf][tid][12] = regs[3]; }

    LOAD_B(0);
    STORE_B(0);
    __syncthreads();

    float minv[8];
    int   mini[8];
    #pragma unroll
    for (int r = 0; r < 8; ++r) { minv[r] = 3.4e38f; mini[r] = 0; }

    const int rbase0 = half << 1;

    for (int t = 0; t < CTILES; ++t) {
        const int buf  = t & 1;
        const int code = nbase + t * 16 + lc;
        const float cn = cnorm[code];               // issue early, used late
        if (t + 1 < CTILES) LOAD_B((t + 1) * 16);   // prefetch next tile

        // ---- stage all 32 B fragments into distinct VGPRs (deep DS pipeline) ----
        v2f bf[32];
        #pragma unroll
        for (int kk = 0; kk < 32; ++kk) {
            const int rb = 4 * kk + rbase0;
            bf[kk].x = Bs[buf][rb][lc];
            bf[kk].y = Bs[buf][rb + 1][lc];
        }

        // ---- 16x16 distance tile over D=128 : 32 fp32 WMMAs, 2 chains ----
        v8f acc0 = {};
        v8f acc1 = {};
        #pragma unroll
        for (int kk = 0; kk < 32; kk += 2) {
            acc0 = __builtin_amdgcn_wmma_f32_16x16x4_f32(
                false, a[kk],     false, bf[kk],     (short)0, acc0, false, false);
            acc1 = __builtin_amdgcn_wmma_f32_16x16x4_f32(
                false, a[kk + 1], false, bf[kk + 1], (short)0, acc1, false, false);
        }

        // dist = ||c||^2 - 2 s.c   (||s||^2 is row-constant -> irrelevant)
        #pragma unroll
        for (int r = 0; r < 8; ++r) {
            float dot  = acc0[r] + acc1[r];
            float dist = __builtin_fmaf(-2.0f, dot, cn);
            if (dist < minv[r]) { minv[r] = dist; mini[r] = code; }
        }

        __syncthreads();
        if (t + 1 < CTILES) {
            STORE_B(buf ^ 1);
            __syncthreads();
        }
    }

    // ---- cross-lane argmin: C-layout row M=r+8*half lives in lanes half*16.. ----
    #pragma unroll
    for (int r = 0; r < 8; ++r) {
        redV[w][r][lane] = minv[r];
        redI[w][r][lane] = mini[r];
    }
    __syncthreads();

    if (lane < 16) {
        const int rr = lane;              // sample row within the wave
        const int r  = rr & 7;
        const int g  = (rr >> 3) * 16;
        float bv = redV[w][r][g];
        int   bi = redI[w][r][g];
        #pragma unroll
        for (int c = 1; c < 16; ++c) {
            float v = redV[w][r][g + c];
            int   i = redI[w][r][g + c];
            if (v < bv || (v == bv && i < bi)) { bv = v; bi = i; }
        }
        const int m = m0 + w * 16 + rr;
        minV[(size_t)chunk * NSAMP + m] = bv;
        minI[(size_t)chunk * NSAMP + m] = bi;
    }
}

// ---------------- combine per-chunk candidates (chunks ascend in code id) ----
__global__ __launch_bounds__(256) void combine_kernel(const float* __restrict__ minV,
                                                      const int* __restrict__ minI,
                                                      int* __restrict__ fidx,
                                                      float* __restrict__ out) {
    int m = blockIdx.x * 256 + threadIdx.x;
    if (m < NSAMP) {
        float bv = minV[m];
        int   bi = minI[m];
        #pragma unroll
        for (int c = 1; c < CHUNKS; ++c) {
            float v = minV[(size_t)c * NSAMP + m];
            int   i = minI[(size_t)c * NSAMP + m];
            if (v < bv) { bv = v; bi = i; }   // strict <: earlier chunk = smaller code
        }
        fidx[m] = bi;
        out[IDX_OFF + m] = (float)bi;         // cluster_index
    }
}

// ---------------- gather quantize = CM[:, idx] + squared-error partial ----
__global__ __launch_bounds__(128) void gather_kernel(const float* __restrict__ S,
                                                     const float* __restrict__ CM,
                                                     const int* __restrict__ fidx,
                                                     float* __restrict__ out,
                                                     float* __restrict__ partial) {
    __shared__ int   sIdx[64];
    __shared__ float redL[128];

    const int tid  = threadIdx.x;
    const int w    = tid >> 5;
    const int lane = tid & 31;
    const int m0   = blockIdx.x * 64;

    if (tid < 64) sIdx[tid] = fidx[m0 + tid];
    __syncthreads();

    float lacc = 0.f;
    for (int s = 0; s < 16; ++s) {
        const int m    = m0 + w * 16 + s;
        const int code = sIdx[w * 16 + s];
        #pragma unroll
        for (int i = 0; i < 4; ++i) {
            const int d = lane + 32 * i;
            float q = CM[(size_t)d * KCODES + code];   // L2-resident codebook
            out[(size_t)m * DIMS + d] = q;
            float df = q - S[(size_t)m * DIMS + d];
            lacc = __builtin_fmaf(df, df, lacc);
        }
    }

    redL[tid] = lacc;
    __syncthreads();
    #pragma unroll
    for (int off = 64; off > 0; off >>= 1) {
        if (tid < off) redL[tid] += redL[tid + off];
        __syncthreads();
    }
    if (tid == 0) partial[blockIdx.x] = redL[0];
}

// ---------------- deterministic final loss reduction ----------------
__global__ __launch_bounds__(256) void loss_kernel(const float* __restrict__ partial,
                                                   float* __restrict__ out) {
    __shared__ float sm[256];
    sm[threadIdx.x] = partial[threadIdx.x];
    __syncthreads();
    #pragma unroll
    for (int off = 128; off > 0; off >>= 1) {
        if (threadIdx.x < off) sm[threadIdx.x] += sm[threadIdx.x + off];
        __syncthreads();
    }
    if (threadIdx.x == 0)
        out[LOSS_OFF] = sm[0] * (1.25f / (float)QSIZE);  // e_loss + 0.25*q_loss
}

extern "C" void kernel_launch(void* const* d_in, const int* in_sizes, int n_in,
                              void* d_out, int out_size, void* d_ws, size_t ws_size,
                              hipStream_t stream) {
    const float* S  = (const float*)d_in[0];   // [16384, 128]
    const float* CM = (const float*)d_in[1];   // [128, 10000]
    float* out      = (float*)d_out;

    // workspace layout (floats): cn | minV | minI | fidx | partial
    float* cn      = (float*)d_ws;                       // 10000 (pad 10240)
    float* minV    = cn + 10240;                         // 5 * 16384
    int*   minI    = (int*)(minV + CHUNKS * NSAMP);      // 5 * 16384
    int*   fidx    = minI + CHUNKS * NSAMP;              // 16384
    float* partial = (float*)(fidx + NSAMP);             // 256

    cnorm_kernel  <<<(KCODES + 255) / 256, 256, 0, stream>>>(CM, cn);
    vq_dist_kernel<<<dim3(NSAMP / 64, CHUNKS), 128, 0, stream>>>(S, CM, cn, minV, minI);
    combine_kernel<<<NSAMP / 256, 256, 0, stream>>>(minV, minI, fidx, out);
    gather_kernel <<<NSAMP / 64, 128, 0, stream>>>(S, CM, fidx, out, partial);
    loss_kernel   <<<1, 256, 0, stream>>>(partial, out);
}